// PKernel_D2_18932215840961
// MI455X (gfx1250) — compile-verified
//
#include <hip/hip_runtime.h>
#include <math.h>

typedef __attribute__((ext_vector_type(2))) float v2f;
typedef __attribute__((ext_vector_type(8))) float v8f;

#define NB   16           // batch
#define NC   64           // channels (IC == INF)
#define HW   65536        // 256*256
#define IMG  256
#define HID  576

__device__ __forceinline__ float sigmoidf_(float x) { return 1.f / (1.f + __expf(-x)); }

// ---------------------------------------------------------------------------
// K1: per-(b,c) spatial avg & max of `weight`  -> avg[1024], mx[1024]
// ---------------------------------------------------------------------------
__global__ void k1_pool(const float* __restrict__ w, float* __restrict__ avg,
                        float* __restrict__ mx) {
    const int bc = blockIdx.x;                 // 0..1023
    const float* p = w + (size_t)bc * HW;
    float s = 0.f, m = -INFINITY;
    for (int i = threadIdx.x; i < HW; i += 256) {
        __builtin_prefetch(p + i + 2048, 0, 1);
        float v = p[i];
        s += v; m = fmaxf(m, v);
    }
    __shared__ float ss[256], sm[256];
    ss[threadIdx.x] = s; sm[threadIdx.x] = m;
    __syncthreads();
    for (int d = 128; d > 0; d >>= 1) {
        if (threadIdx.x < d) {
            ss[threadIdx.x] += ss[threadIdx.x + d];
            sm[threadIdx.x] = fmaxf(sm[threadIdx.x], sm[threadIdx.x + d]);
        }
        __syncthreads();
    }
    if (threadIdx.x == 0) { avg[bc] = ss[0] * (1.f / (float)HW); mx[bc] = sm[0]; }
}

// ---------------------------------------------------------------------------
// K2: channel-attention MLP via V_WMMA_F32_16X16X4_F32 (one wave32)
//   ca = sigmoid( (relu(avg@fc1^T) + relu(mx@fc1^T)) @ fc2^T )   [16,64]
// A frag (16x4):  a[v] = A[lane%16][k0 + 2*(lane/16) + v]
// B frag (4x16):  b[v] = B[k0 + 2*(lane/16) + v][lane%16]
// C/D (16x16):    c[i] -> row i + 8*(lane/16), col lane%16
// B1 (fc1^T, zero-padded to 16 cols) staged in LDS to avoid divergent loads.
// ---------------------------------------------------------------------------
__global__ void k2_ca(const float* __restrict__ avg, const float* __restrict__ mx,
                      const float* __restrict__ fc1, const float* __restrict__ fc2,
                      float* __restrict__ ca) {
    const int l  = threadIdx.x;        // 0..31
    const int m  = l & 15;
    const int kh = (l >> 4) * 2;       // 0 or 2
    __shared__ float Bf1[NC][16];      // Bf1[k][n] = fc1[n][k] (n<8), else 0
    __shared__ float hsum[16][16];

    for (int i = l; i < NC * 16; i += 32) {
        int k = i >> 4, n = i & 15;
        Bf1[k][n] = (n < 8) ? fc1[n * NC + k] : 0.f;
    }
    __syncthreads();

    v8f c_avg = {}; v8f c_mx = {};
    for (int kk = 0; kk < NC; kk += 4) {
        v2f a0, a1, b0;
        a0.x = avg[m * NC + kk + kh + 0];
        a0.y = avg[m * NC + kk + kh + 1];
        a1.x = mx [m * NC + kk + kh + 0];
        a1.y = mx [m * NC + kk + kh + 1];
        b0.x = Bf1[kk + kh + 0][m];
        b0.y = Bf1[kk + kh + 1][m];
        c_avg = __builtin_amdgcn_wmma_f32_16x16x4_f32(false, a0, false, b0,
                                                      (short)0, c_avg, false, false);
        c_mx  = __builtin_amdgcn_wmma_f32_16x16x4_f32(false, a1, false, b0,
                                                      (short)0, c_mx, false, false);
    }
    for (int i = 0; i < 8; ++i) {
        int row = i + (l >> 4) * 8;
        hsum[row][m] = fmaxf(c_avg[i], 0.f) + fmaxf(c_mx[i], 0.f);
    }
    __syncthreads();

    // second layer: N in 4 tiles of 16, K = 8 (hidden dim)
    for (int t = 0; t < 4; ++t) {
        v8f c2 = {};
        for (int kk = 0; kk < 8; kk += 4) {
            v2f a, b;
            a.x = hsum[m][kk + kh + 0];
            a.y = hsum[m][kk + kh + 1];
            int n = t * 16 + m;                 // fc2 is [64,8]
            b.x = fc2[n * 8 + kk + kh + 0];
            b.y = fc2[n * 8 + kk + kh + 1];
            c2 = __builtin_amdgcn_wmma_f32_16x16x4_f32(false, a, false, b,
                                                       (short)0, c2, false, false);
        }
        for (int i = 0; i < 8; ++i) {
            int row = i + (l >> 4) * 8;
            ca[row * NC + t * 16 + m] = sigmoidf_(c2[i]);
        }
    }
}

// ---------------------------------------------------------------------------
// K3: per-pixel channel mean & max of wg = ca*weight  -> spmean, spmax [16,HW]
// ---------------------------------------------------------------------------
__global__ void k3_chanstats(const float* __restrict__ w, const float* __restrict__ ca,
                             float* __restrict__ spmean, float* __restrict__ spmax) {
    const int b = blockIdx.y;
    const int pix = blockIdx.x * 256 + threadIdx.x;      // gridDim.x = 256
    __shared__ float cab[NC];
    if (threadIdx.x < NC) cab[threadIdx.x] = ca[b * NC + threadIdx.x];
    __syncthreads();
    const float* pw = w + (size_t)b * NC * HW + pix;
    float s = 0.f, m = -INFINITY;
    #pragma unroll 4
    for (int c = 0; c < NC; ++c) {
        float v = cab[c] * pw[(size_t)c * HW];
        s += v; m = fmaxf(m, v);
    }
    spmean[b * HW + pix] = s * (1.f / (float)NC);
    spmax [b * HW + pix] = m;
}

// ---------------------------------------------------------------------------
// K4: 7x7 conv (pad 3) over [mean,max] maps + sigmoid -> sa [16,HW]
//     LDS-tiled: 16x16 output tile, 22x22 halo tiles
// ---------------------------------------------------------------------------
__global__ void k4_sa(const float* __restrict__ spmean, const float* __restrict__ spmax,
                      const float* __restrict__ saw, float* __restrict__ sa) {
    const int b   = blockIdx.z;
    const int tx0 = blockIdx.x * 16, ty0 = blockIdx.y * 16;
    __shared__ float t0[22][22], t1[22][22];
    __shared__ float wsm[98];
    for (int idx = threadIdx.x; idx < 22 * 22; idx += 256) {
        int ly = idx / 22, lx = idx % 22;
        int gy = ty0 + ly - 3, gx = tx0 + lx - 3;
        bool ok = (gy >= 0) & (gy < IMG) & (gx >= 0) & (gx < IMG);
        t0[ly][lx] = ok ? spmean[b * HW + gy * IMG + gx] : 0.f;
        t1[ly][lx] = ok ? spmax [b * HW + gy * IMG + gx] : 0.f;
    }
    if (threadIdx.x < 98) wsm[threadIdx.x] = saw[threadIdx.x];
    __syncthreads();
    const int ly = threadIdx.x / 16, lx = threadIdx.x % 16;
    float acc = 0.f;
    #pragma unroll
    for (int dy = 0; dy < 7; ++dy)
        #pragma unroll
        for (int dx = 0; dx < 7; ++dx) {
            acc += t0[ly + dy][lx + dx] * wsm[dy * 7 + dx];
            acc += t1[ly + dy][lx + dx] * wsm[49 + dy * 7 + dx];
        }
    sa[b * HW + (ty0 + ly) * IMG + (tx0 + lx)] = sigmoidf_(acc);
}

// ---------------------------------------------------------------------------
// K5: swsum[b,c] = sum_hw sa[b,hw] * weight[b,c,hw]
// ---------------------------------------------------------------------------
__global__ void k5_swsum(const float* __restrict__ w, const float* __restrict__ sa,
                         float* __restrict__ swsum) {
    const int bc = blockIdx.x;
    const int b  = bc >> 6;
    const float* pw = w + (size_t)bc * HW;
    const float* ps = sa + (size_t)b * HW;
    float s = 0.f;
    for (int i = threadIdx.x; i < HW; i += 256) {
        __builtin_prefetch(pw + i + 2048, 0, 1);
        s += pw[i] * ps[i];
    }
    __shared__ float ss[256];
    ss[threadIdx.x] = s;
    __syncthreads();
    for (int d = 128; d > 0; d >>= 1) {
        if (threadIdx.x < d) ss[threadIdx.x] += ss[threadIdx.x + d];
        __syncthreads();
    }
    if (threadIdx.x == 0) swsum[bc] = ss[0];
}

// ---------------------------------------------------------------------------
// K6: pooled = (ca*swsum/HW) @ W1^T + b1 via WMMA, then per-sample 3x3 kernels
//     kern[b, j] = pooled[b, j/9] * w2[j] + b2[j],  j < 576   (one wave32)
// ---------------------------------------------------------------------------
__global__ void k6_kern(const float* __restrict__ ca, const float* __restrict__ swsum,
                        const float* __restrict__ w1, const float* __restrict__ b1,
                        const float* __restrict__ w2, const float* __restrict__ b2,
                        float* __restrict__ kern) {
    const int l  = threadIdx.x;
    const int m  = l & 15;
    const int kh = (l >> 4) * 2;
    __shared__ float mw[16][NC];
    __shared__ float pooled[16][NC];
    for (int i = l; i < 16 * NC; i += 32)
        mw[i >> 6][i & 63] = ca[i] * swsum[i] * (1.f / (float)HW);
    __syncthreads();
    for (int t = 0; t < 4; ++t) {
        v8f c = {};
        for (int kk = 0; kk < NC; kk += 4) {
            v2f a, b;
            a.x = mw[m][kk + kh + 0];
            a.y = mw[m][kk + kh + 1];
            int n = t * 16 + m;                 // w1 is [64,64]: out[o] = sum_c mw[c]*w1[o][c]
            b.x = w1[n * NC + kk + kh + 0];
            b.y = w1[n * NC + kk + kh + 1];
            c = __builtin_amdgcn_wmma_f32_16x16x4_f32(false, a, false, b,
                                                      (short)0, c, false, false);
        }
        for (int i = 0; i < 8; ++i) {
            int row = i + (l >> 4) * 8, col = t * 16 + m;
            pooled[row][col] = c[i] + b1[col];
        }
    }
    __syncthreads();
    for (int i = l; i < NB * HID; i += 32) {
        int b = i / HID, j = i % HID;
        kern[i] = pooled[b][j / 9] * w2[j] + b2[j];
    }
}

// ---------------------------------------------------------------------------
// K7: dynamic per-sample 3x3 conv (sum over 64 channels), pad 1
//     32x32 output tile, 34x34 LDS halo tile per channel (1.13x halo amp),
//     256 threads, 4 output pixels per thread (rows ly, ly+8, ly+16, ly+24)
// ---------------------------------------------------------------------------
__global__ void k7_dynconv(const float* __restrict__ x, const float* __restrict__ kern,
                           float* __restrict__ out) {
    const int b   = blockIdx.z;
    const int tx0 = blockIdx.x * 32, ty0 = blockIdx.y * 32;
    __shared__ float kw[HID];
    __shared__ float tile[34][34];
    for (int i = threadIdx.x; i < HID; i += 256) kw[i] = kern[b * HID + i];
    const int lx = threadIdx.x & 31;       // 0..31
    const int ly = threadIdx.x >> 5;       // 0..7
    float acc0 = 0.f, acc1 = 0.f, acc2 = 0.f, acc3 = 0.f;
    for (int c = 0; c < NC; ++c) {
        __syncthreads();
        const float* px = x + ((size_t)b * NC + c) * HW;
        for (int i = threadIdx.x; i < 34 * 34; i += 256) {
            int y = i / 34, xc = i % 34;
            int gy = ty0 + y - 1, gx = tx0 + xc - 1;
            bool ok = (gy >= 0) & (gy < IMG) & (gx >= 0) & (gx < IMG);
            tile[y][xc] = ok ? px[gy * IMG + gx] : 0.f;
        }
        __syncthreads();
        const float* kc = &kw[c * 9];
        const float k0 = kc[0], k1 = kc[1], k2 = kc[2],
                    k3 = kc[3], k4 = kc[4], k5 = kc[5],
                    k6 = kc[6], k7 = kc[7], k8 = kc[8];
        #pragma unroll
        for (int r = 0; r < 4; ++r) {
            const int yy = ly + 8 * r;
            float v = tile[yy + 0][lx + 0] * k0 + tile[yy + 0][lx + 1] * k1 + tile[yy + 0][lx + 2] * k2
                    + tile[yy + 1][lx + 0] * k3 + tile[yy + 1][lx + 1] * k4 + tile[yy + 1][lx + 2] * k5
                    + tile[yy + 2][lx + 0] * k6 + tile[yy + 2][lx + 1] * k7 + tile[yy + 2][lx + 2] * k8;
            if (r == 0) acc0 += v; else if (r == 1) acc1 += v;
            else if (r == 2) acc2 += v; else acc3 += v;
        }
    }
    out[(size_t)b * HW + (ty0 + ly +  0) * IMG + (tx0 + lx)] = acc0;
    out[(size_t)b * HW + (ty0 + ly +  8) * IMG + (tx0 + lx)] = acc1;
    out[(size_t)b * HW + (ty0 + ly + 16) * IMG + (tx0 + lx)] = acc2;
    out[(size_t)b * HW + (ty0 + ly + 24) * IMG + (tx0 + lx)] = acc3;
}

// ---------------------------------------------------------------------------
extern "C" void kernel_launch(void* const* d_in, const int* in_sizes, int n_in,
                              void* d_out, int out_size, void* d_ws, size_t ws_size,
                              hipStream_t stream) {
    const float* x1   = (const float*)d_in[0];   // [16,64,256,256]
    const float* wgt  = (const float*)d_in[1];   // [16,64,256,256]
    const float* fc1  = (const float*)d_in[2];   // [8,64]
    const float* fc2  = (const float*)d_in[3];   // [64,8]
    const float* saw  = (const float*)d_in[4];   // [1,2,7,7]
    const float* w1   = (const float*)d_in[5];   // [64,64]
    const float* b1   = (const float*)d_in[6];   // [64]
    const float* w2   = (const float*)d_in[7];   // [576]
    const float* b2   = (const float*)d_in[8];   // [576]
    float* out = (float*)d_out;                  // [16,1,256,256]

    float* ws = (float*)d_ws;
    // workspace layout (floats); total ~= 12.1 MB
    float* avg    = ws + 0;                 // 1024
    float* mx     = ws + 1024;              // 1024
    float* ca     = ws + 2048;              // 1024
    float* swsum  = ws + 3072;              // 1024
    float* kern   = ws + 4096;              // 9216
    float* spmean = ws + 16384;             // 1048576
    float* spmax  = spmean + 1048576;       // 1048576
    float* sa     = spmax  + 1048576;       // 1048576

    k1_pool<<<dim3(NB * NC), dim3(256), 0, stream>>>(wgt, avg, mx);
    k2_ca<<<dim3(1), dim3(32), 0, stream>>>(avg, mx, fc1, fc2, ca);
    k3_chanstats<<<dim3(HW / 256, NB), dim3(256), 0, stream>>>(wgt, ca, spmean, spmax);
    k4_sa<<<dim3(IMG / 16, IMG / 16, NB), dim3(256), 0, stream>>>(spmean, spmax, saw, sa);
    k5_swsum<<<dim3(NB * NC), dim3(256), 0, stream>>>(wgt, sa, swsum);
    k6_kern<<<dim3(1), dim3(32), 0, stream>>>(ca, swsum, w1, b1, w2, b2, kern);
    k7_dynconv<<<dim3(IMG / 32, IMG / 32, NB), dim3(256), 0, stream>>>(x1, kern, out);
}